// GCN_29265907155090
// MI455X (gfx1250) — compile-verified
//
#include <hip/hip_runtime.h>

typedef __attribute__((ext_vector_type(2))) float v2f;
typedef __attribute__((ext_vector_type(8))) float v8f;

#define FEAT 64

// ---------------------------------------------------------------------------
// Edge kernel: msg = h[src] * w ; agg[dst] += msg   (64 threads per edge)
// Consecutive lanes cover consecutive features -> coalesced 128B/wave reads,
// atomics land in the 192MB L2 (whole agg buffer is L2-resident).
// ---------------------------------------------------------------------------
__global__ void __launch_bounds__(256)
gcn_scatter(const float* __restrict__ h,
            const float* __restrict__ ew,
            const int*   __restrict__ src,
            const int*   __restrict__ dst,
            float*       __restrict__ agg,
            int n_edges) {
    int idx = blockIdx.x * 256 + threadIdx.x;
    int e = idx >> 6;
    if (e >= n_edges) return;
    int f = idx & 63;
    int s = src[e];
    int d = dst[e];
    float w = ew[e];
    float val = h[(size_t)s * FEAT + f] * w;
    atomicAdd(&agg[(size_t)d * FEAT + f], val);
}

// ---------------------------------------------------------------------------
// Dense part: out = relu(agg @ W + b), W is 64x64 f32 staged in LDS.
// One wave per 16-row strip; 4 column tiles x 16 K-steps of
// V_WMMA_F32_16X16X4_F32 (f32 in / f32 out -> matches reference precision).
//
// Fragment layouts (wave32, ISA 7.12.2):
//   A 16x4 : lanes 0-15 -> M=lane,  VGPR{0,1}=K{0,1}; lanes 16-31 -> K{2,3}
//   B 4x16 : lane&15 = N, (2*(lane>>4) + VGPR) = K
//   C 16x16: VGPR v, lanes 0-15 -> M=v, N=lane; lanes 16-31 -> M=v+8
// ---------------------------------------------------------------------------
__global__ void __launch_bounds__(256)
gcn_gemm_relu(const float* __restrict__ agg,
              const float* __restrict__ W,      // [64,64] row-major (k,n)
              const float* __restrict__ bias,   // [64]
              float*       __restrict__ out,
              int n_rows) {
    __shared__ float Wlds[FEAT * FEAT];

    // cooperative load of W into LDS (16KB), float4 granularity
    for (int i = threadIdx.x; i < FEAT * FEAT / 4; i += 256) {
        ((float4*)Wlds)[i] = ((const float4*)W)[i];
    }
    __syncthreads();

    const int lane = threadIdx.x & 31;
    const int wave = threadIdx.x >> 5;
    const int tile = blockIdx.x * 8 + wave;          // 16-row strip index
    if (tile * 16 >= n_rows) return;                 // wave-uniform guard

    const int nlo = lane & 15;   // A row / B,C column within tile
    const int hi  = lane >> 4;   // lane half selects K offset (A/B) or M+8 (C)

    const float* aRow = agg + (size_t)(tile * 16 + nlo) * FEAT;

    v8f zero = {};
    v8f acc[4] = {zero, zero, zero, zero};

    for (int k0 = 0; k0 < FEAT; k0 += 4) {
        const int ka = k0 + 2 * hi;
        v2f a;
        a.x = aRow[ka];
        a.y = aRow[ka + 1];
        #pragma unroll
        for (int c = 0; c < 4; ++c) {
            v2f b;
            b.x = Wlds[ka * FEAT + c * 16 + nlo];
            b.y = Wlds[(ka + 1) * FEAT + c * 16 + nlo];
            acc[c] = __builtin_amdgcn_wmma_f32_16x16x4_f32(
                /*neg_a=*/false, a, /*neg_b=*/false, b,
                /*c_mod=*/(short)0, acc[c],
                /*reuse_a=*/false, /*reuse_b=*/false);
        }
    }

    // bias + relu + store (C/D layout: VGPR v -> row v (+8 for upper lanes))
    #pragma unroll
    for (int c = 0; c < 4; ++c) {
        const int col = c * 16 + nlo;
        const float bv = bias[col];
        #pragma unroll
        for (int v = 0; v < 8; ++v) {
            const int r = tile * 16 + v + 8 * hi;
            float val = acc[c][v] + bv;
            out[(size_t)r * FEAT + col] = val > 0.0f ? val : 0.0f;
        }
    }
}

// ---------------------------------------------------------------------------
// Driver: 2 GCN layers. agg and intermediate h1 live in workspace.
// ---------------------------------------------------------------------------
extern "C" void kernel_launch(void* const* d_in, const int* in_sizes, int n_in,
                              void* d_out, int out_size, void* d_ws, size_t ws_size,
                              hipStream_t stream) {
    const float* h   = (const float*)d_in[0];   // [N,64]
    const float* Ws  = (const float*)d_in[1];   // [2,64,64]
    const float* bs  = (const float*)d_in[2];   // [2,64]
    const float* ew  = (const float*)d_in[3];   // [E]
    const int*   src = (const int*)d_in[4];     // [E]
    const int*   dst = (const int*)d_in[5];     // [E]
    float* out = (float*)d_out;

    const int N = in_sizes[0] / FEAT;   // 50000
    const int E = in_sizes[3];          // 800000

    float* agg = (float*)d_ws;
    float* h1  = agg + (size_t)N * FEAT;

    const size_t aggBytes = (size_t)N * FEAT * sizeof(float);

    const int scatterThreads = E * FEAT;                 // 51.2M
    dim3 sGrd((scatterThreads + 255) / 256), sBlk(256);

    const int tiles = (N + 15) / 16;                     // 3125
    dim3 gGrd((tiles + 7) / 8), gBlk(256);               // 8 waves/block

    // ---- layer 0 ----
    hipMemsetAsync(agg, 0, aggBytes, stream);
    gcn_scatter<<<sGrd, sBlk, 0, stream>>>(h, ew, src, dst, agg, E);
    gcn_gemm_relu<<<gGrd, gBlk, 0, stream>>>(agg, Ws, bs, h1, N);

    // ---- layer 1 ----
    hipMemsetAsync(agg, 0, aggBytes, stream);
    gcn_scatter<<<sGrd, sBlk, 0, stream>>>(h1, ew, src, dst, agg, E);
    gcn_gemm_relu<<<gGrd, gBlk, 0, stream>>>(agg, Ws + FEAT * FEAT, bs + FEAT, out, N);
}